// Seq2Seq_18227841204411
// MI455X (gfx1250) — compile-verified
//
#include <hip/hip_runtime.h>
#include <math.h>

#define HID   1024
#define H3    3072
#define SEQ_E 4096
#define SEQ_D 2048
#define NVOC  256
#define NB    64          // persistent workgroups (must all be resident)
#define TPB   256
#define SLICE 16          // h-dims owned per WG in GRU recurrence
#define START_TOK 86

typedef float v2f __attribute__((ext_vector_type(2)));
typedef float v8f __attribute__((ext_vector_type(8)));

#if defined(__HIP_DEVICE_COMPILE__) && __has_builtin(__builtin_amdgcn_tensor_load_to_lds)
#define HAVE_TDM 1
#else
#define HAVE_TDM 0
#endif

__device__ __forceinline__ float sigm(float x) { return 1.0f / (1.0f + expf(-x)); }

__device__ __forceinline__ float warp_red_add(float v) {
#pragma unroll
  for (int m = 16; m >= 1; m >>= 1) v += __shfl_xor(v, m, 32);
  return v;
}

// One wave32 computes dot(row[0:1024], vec[0:1024]); result valid in all lanes.
__device__ __forceinline__ float dot1024(const float* __restrict__ row,
                                         const float* __restrict__ vec,
                                         int lane) {
  float acc = 0.f;
#pragma unroll
  for (int k4 = 0; k4 < 8; ++k4) {
    int k = (k4 * 32 + lane) * 4;
    float4 w = *(const float4*)(row + k);
    float4 h = *(const float4*)(vec + k);
    acc = fmaf(w.x, h.x, acc);
    acc = fmaf(w.y, h.y, acc);
    acc = fmaf(w.z, h.z, acc);
    acc = fmaf(w.w, h.w, acc);
  }
  return warp_red_add(acc);
}

// Device-wide sense-reversing barrier. bar[0]=counter (must start 0), bar[1]=generation.
__device__ void grid_barrier(unsigned* bar, unsigned nb) {
  __syncthreads();
  if (threadIdx.x == 0) {
    __threadfence();
    volatile unsigned* cnt = bar;
    volatile unsigned* gen = bar + 1;
    unsigned g = *gen;
    if (atomicAdd(bar, 1u) == nb - 1u) {
      *cnt = 0u;
      __threadfence();
      atomicAdd(bar + 1, 1u);
    } else {
      while (*gen == g) { __builtin_amdgcn_s_sleep(2); }
    }
    __threadfence();
  }
  __syncthreads();
}

#if HAVE_TDM
typedef unsigned int u32x4 __attribute__((ext_vector_type(4)));
typedef int          i32x8 __attribute__((ext_vector_type(8)));
typedef int          i32x4 __attribute__((ext_vector_type(4)));
typedef __attribute__((address_space(3))) float lds_f;

__device__ __forceinline__ unsigned lds_byte_off(void* p) {
  return (unsigned)(unsigned long long)(lds_f*)p;   // generic -> AS3 -> LDS byte offset
}

// Issue one TDM 2D tile load: rows x row_elems f32, global row pitch = pitch_elems.
// D# layout per CDNA5 ISA ch.8 (group0: count/lds_addr/global_addr/type,
// group1: data_size/tensor dims/tile dims/strides). 2D: groups 2-3 unused (zero).
// This toolchain's builtin is the 6-arg (clang-23 / therock-10.0) form:
//   (u32x4 g0, i32x8 g1, i32x4 g2, i32x4 g3, i32x8 ext, i32 cpol)
__device__ __forceinline__ void tdm_load_2d(unsigned lds_byte, const void* gptr,
                                            unsigned rows, unsigned row_elems,
                                            unsigned pitch_elems) {
  unsigned long long ga = (unsigned long long)(uintptr_t)gptr;
  u32x4 g0;
  g0.x = 1u;                                            // count=1 (user descriptor)
  g0.y = lds_byte;                                      // lds_addr
  g0.z = (unsigned)(ga & 0xFFFFFFFFu);                  // global_addr[31:0]
  g0.w = (unsigned)((ga >> 32) & 0x01FFFFFFu) | (2u << 30);  // global_addr[56:32] | type=2
  i32x8 g1;
  g1[0] = (int)(2u << 16);                                    // data_size=4B
  g1[1] = (int)((row_elems & 0xFFFFu) << 16);                 // tensor_dim0[15:0]
  g1[2] = (int)(((row_elems >> 16) & 0xFFFFu) |               // tensor_dim0[31:16]
                ((rows & 0xFFFFu) << 16));                    // tensor_dim1[15:0]
  g1[3] = (int)(((rows >> 16) & 0xFFFFu) |                    // tensor_dim1[31:16]
                ((row_elems & 0xFFFFu) << 16));               // tile_dim0
  g1[4] = (int)(rows & 0xFFFFu);                              // tile_dim1 (tile_dim2=0)
  g1[5] = (int)pitch_elems;                                   // tensor_dim0_stride[31:0]
  g1[6] = 0;
  g1[7] = 0;
  i32x4 z4 = {0, 0, 0, 0};
  i32x8 z8 = {0, 0, 0, 0, 0, 0, 0, 0};
  __builtin_amdgcn_tensor_load_to_lds(g0, g1, z4, z4, z8, 0);
}
#endif

// Stage 48 Whh rows (gates r,z,n of h-dims [j0, j0+16)) into Wsh (192 KB LDS).
__device__ __forceinline__ void stage_whh_slice(float* Wsh, const float* Whh, int j0,
                                                int tid) {
#if HAVE_TDM
  if (tid == 0) {
    unsigned base = lds_byte_off(Wsh);
#pragma unroll
    for (int g = 0; g < 3; ++g)
      tdm_load_2d(base + (unsigned)g * SLICE * HID * 4u,
                  Whh + ((size_t)g * HID + j0) * HID, SLICE, HID, HID);
    __builtin_amdgcn_s_wait_tensorcnt(0);
  }
  __syncthreads();
#else
  for (int idx = tid; idx < 3 * SLICE * (HID / 4); idx += TPB) {
    const int lr = idx >> 8;              // local row (HID/4 == 256 float4/row)
    const int c4 = idx & 255;
    const int gr = (lr / SLICE) * HID + j0 + (lr % SLICE);
    ((float4*)Wsh)[lr * (HID / 4) + c4] =
        ((const float4*)(Whh + (size_t)gr * HID))[c4];
  }
  __syncthreads();
#endif
}

// ---------------------------------------------------------------------------
// C[S x 3072] = A[S x 1024] @ W[3072 x 1024]^T + bias, f32 WMMA 16x16x4.
// Optional fused embedding gather: A row m = emb[gather[m]].
// Wave (wm,wn) of an 8-wave block owns a 16(M) x 64(N) tile (4 accumulators).
// ---------------------------------------------------------------------------
__global__ __launch_bounds__(TPB) void wmma_gemm_xwT_bias(
    const float* __restrict__ A, const int* __restrict__ gather,
    const float* __restrict__ W, const float* __restrict__ bias,
    float* __restrict__ C, int S) {
  const int tid  = threadIdx.x;
  const int lane = tid & 31;
  const int wave = tid >> 5;
  const int wm = wave & 1, wn = wave >> 1;
  const int m0 = blockIdx.x * 32 + wm * 16;
  const int n0 = blockIdx.y * 256 + wn * 64;
  const int l15 = lane & 15;
  const int khalf = (lane >> 4) * 2;  // lanes 0-15: K+0..1, lanes 16-31: K+2..3

  const int arowi = m0 + l15;
  const float* arow = gather ? (A + (size_t)gather[arowi] * HID)
                             : (A + (size_t)arowi * HID);
  const float* brow0 = W + (size_t)(n0 + 0 * 16 + l15) * HID;
  const float* brow1 = W + (size_t)(n0 + 1 * 16 + l15) * HID;
  const float* brow2 = W + (size_t)(n0 + 2 * 16 + l15) * HID;
  const float* brow3 = W + (size_t)(n0 + 3 * 16 + l15) * HID;

  v8f acc0 = {0.f, 0.f, 0.f, 0.f, 0.f, 0.f, 0.f, 0.f};
  v8f acc1 = acc0, acc2 = acc0, acc3 = acc0;

#pragma unroll 4
  for (int k = 0; k < HID; k += 4) {
    v2f a  = *(const v2f*)(arow  + k + khalf);
    v2f b0 = *(const v2f*)(brow0 + k + khalf);
    v2f b1 = *(const v2f*)(brow1 + k + khalf);
    v2f b2 = *(const v2f*)(brow2 + k + khalf);
    v2f b3 = *(const v2f*)(brow3 + k + khalf);
    acc0 = __builtin_amdgcn_wmma_f32_16x16x4_f32(false, a, false, b0, (short)0, acc0, false, false);
    acc1 = __builtin_amdgcn_wmma_f32_16x16x4_f32(false, a, false, b1, (short)0, acc1, false, false);
    acc2 = __builtin_amdgcn_wmma_f32_16x16x4_f32(false, a, false, b2, (short)0, acc2, false, false);
    acc3 = __builtin_amdgcn_wmma_f32_16x16x4_f32(false, a, false, b3, (short)0, acc3, false, false);
  }

  const int hi = lane >> 4;
  v8f accs[4] = {acc0, acc1, acc2, acc3};
#pragma unroll
  for (int s = 0; s < 4; ++s) {
    const int ncol = n0 + s * 16 + l15;
    const float bv = bias[ncol];
#pragma unroll
    for (int v = 0; v < 8; ++v) {
      const int mr = m0 + v + 8 * hi;     // C/D layout: VGPR v = rows v, v+8
      C[(size_t)mr * H3 + ncol] = accs[s][v] + bv;
    }
  }
}

// ---------------------------------------------------------------------------
// Persistent GRU recurrence over S steps. Gi (S x 3072) has bih included.
// Each of NB=64 WGs owns SLICE=16 h-dims; its 48 Whh rows (192 KB) live in LDS.
// Y[t] doubles as h_{t}; h broadcast via global (L2) + grid barrier per step.
// ---------------------------------------------------------------------------
__global__ __launch_bounds__(TPB) void gru_layer_seq(
    const float* __restrict__ Gi, const float* __restrict__ Whh,
    const float* __restrict__ bhh, const float* __restrict__ h_init,
    float* __restrict__ Y, int S, unsigned* bar) {
  __shared__ __align__(16) float Wsh[3 * SLICE * HID];  // 192 KB CDNA5 LDS
  __shared__ __align__(16) float hsh[HID];
  __shared__ float gsh[3 * SLICE];

  const int tid = threadIdx.x, lane = tid & 31, wave = tid >> 5;
  const int j0 = blockIdx.x * SLICE;

  stage_whh_slice(Wsh, Whh, j0, tid);

  for (int t = 0; t < S; ++t) {
    const float* hprev = (t == 0) ? h_init : (Y + (size_t)(t - 1) * HID);
    for (int i = tid; i < HID; i += TPB) hsh[i] = hprev[i];
    if (tid == 0)  // warm next step's precomputed gates while we stall on barrier
      __builtin_prefetch(Gi + (size_t)(t + 1) * H3 + j0, 0, 1);
    __syncthreads();

    for (int d = wave; d < 3 * SLICE; d += 8) {  // 48 dots, 6 per wave
      const float dp = dot1024(Wsh + (size_t)d * HID, hsh, lane);
      if (lane == 0) gsh[d] = dp + bhh[(d / SLICE) * HID + j0 + (d % SLICE)];
    }
    __syncthreads();

    if (tid < SLICE) {
      const int j = j0 + tid;
      const float* gi = Gi + (size_t)t * H3;
      const float r = sigm(gi[j] + gsh[tid]);
      const float z = sigm(gi[HID + j] + gsh[SLICE + tid]);
      const float n = tanhf(gi[2 * HID + j] + r * gsh[2 * SLICE + tid]);
      Y[(size_t)t * HID + j] = (1.f - z) * n + z * hsh[j];
    }
    grid_barrier(bar, NB);
  }
}

// ---------------------------------------------------------------------------
// Persistent decoder: 6 phases per step, separated by 5 barriers.
//   P1 layer-0 GRU (Whh0 slice in LDS)           -> h0buf[nxt]    | barrier
//   P2 fused gi1 + layer-1 GRU (rows = own slice) -> h1buf[nxt]   | barrier
//   P3 attention scores + per-WG max              -> redA         | barrier
//   P4 global max, exp, per-WG sum                -> wts, redB    | barrier
//   P5 context vector                             -> ctx          | barrier
//   P6 output projection                          -> out          | (none)
// ---------------------------------------------------------------------------
__global__ __launch_bounds__(TPB) void decoder_seq(
    const float* __restrict__ Gi0,   // SEQ_D x 3072 (dbih0 included)
    const float* __restrict__ Whh0, const float* __restrict__ bhh0,
    const float* __restrict__ Wih1, const float* __restrict__ bih1,
    const float* __restrict__ Whh1, const float* __restrict__ bhh1,
    const float* __restrict__ enc_out,  // SEQ_E x 1024
    const float* eh0, const float* eh1, // encoder final hiddens (1024 each)
    const float* __restrict__ projW,    // 256 x 2048
    const float* __restrict__ projB,    // 256
    float* __restrict__ h0buf, float* __restrict__ h1buf,  // [2][1024]
    float* __restrict__ wts,            // 4096
    float* __restrict__ redA, float* __restrict__ redB,    // 64 each
    float* __restrict__ ctx,            // 1024
    float* __restrict__ out,            // SEQ_D x 256
    unsigned* bar) {
  __shared__ __align__(16) float Wsh[3 * SLICE * HID];  // 192 KB: Whh0 slice
  __shared__ __align__(16) float sh[SEQ_E];             // 16 KB multi-purpose
  __shared__ float gsh[64];
  __shared__ float gish[3 * SLICE];
  __shared__ float rsh[TPB];
  __shared__ float bcast;

  const int tid = threadIdx.x, lane = tid & 31, wave = tid >> 5;
  const int bid = blockIdx.x;
  const int j0 = bid * SLICE;

  stage_whh_slice(Wsh, Whh0, j0, tid);

  for (int i = bid * TPB + tid; i < HID; i += NB * TPB) {
    h0buf[i] = eh0[i];
    h1buf[i] = eh1[i];
  }
  grid_barrier(bar, NB);

  for (int t = 0; t < SEQ_D; ++t) {
    const int cur = t & 1, nxt = cur ^ 1;

    // P1: layer-0 GRU step (weights in LDS)
    {
      const float* h0c = h0buf + cur * HID;
      for (int i = tid; i < HID; i += TPB) sh[i] = h0c[i];
      if (tid == 0)
        __builtin_prefetch(Gi0 + (size_t)(t + 1) * H3 + j0, 0, 1);
      __syncthreads();
      for (int d = wave; d < 3 * SLICE; d += 8) {
        const float dp = dot1024(Wsh + (size_t)d * HID, sh, lane);
        if (lane == 0) gsh[d] = dp + bhh0[(d / SLICE) * HID + j0 + (d % SLICE)];
      }
      __syncthreads();
      if (tid < SLICE) {
        const int j = j0 + tid;
        const float* gi = Gi0 + (size_t)t * H3;
        const float r = sigm(gi[j] + gsh[tid]);
        const float z = sigm(gi[HID + j] + gsh[SLICE + tid]);
        const float n = tanhf(gi[2 * HID + j] + r * gsh[2 * SLICE + tid]);
        h0buf[nxt * HID + j] = (1.f - z) * n + z * sh[j];
      }
    }
    grid_barrier(bar, NB);

    // P2: fused  gi1 = Wih1 @ h0_new + bih1  and  layer-1 GRU step.
    // Each WG computes exactly the 48 gi1 rows its own h-slice consumes.
    {
      const float* x1  = h0buf + nxt * HID;
      const float* h1c = h1buf + cur * HID;
      for (int i = tid; i < HID; i += TPB) { sh[i] = x1[i]; sh[HID + i] = h1c[i]; }
      __syncthreads();
      for (int d = wave; d < 3 * SLICE; d += 8) {
        const int grow = (d / SLICE) * HID + j0 + (d % SLICE);
        const float gi = dot1024(Wih1 + (size_t)grow * HID, sh, lane);
        const float gh = dot1024(Whh1 + (size_t)grow * HID, sh + HID, lane);
        if (lane == 0) { gish[d] = gi + bih1[grow]; gsh[d] = gh + bhh1[grow]; }
      }
      __syncthreads();
      if (tid < SLICE) {
        const int j = j0 + tid;
        const float r = sigm(gish[tid] + gsh[tid]);
        const float z = sigm(gish[SLICE + tid] + gsh[SLICE + tid]);
        const float n = tanhf(gish[2 * SLICE + tid] + r * gsh[2 * SLICE + tid]);
        h1buf[nxt * HID + j] = (1.f - z) * n + z * sh[HID + j];
      }
    }
    grid_barrier(bar, NB);

    // P3: attention scores (64 rows of enc_out per WG) + per-WG max
    {
      const float* h1n = h1buf + nxt * HID;
      for (int i = tid; i < HID; i += TPB) sh[i] = h1n[i];
      __syncthreads();
      const int s0 = bid * 64;
      for (int d = wave; d < 64; d += 8) {
        const float dp = dot1024(enc_out + (size_t)(s0 + d) * HID, sh, lane);
        if (lane == 0) gsh[d] = dp;
      }
      __syncthreads();
      if (tid < 32) {
        float m = fmaxf(gsh[tid], gsh[tid + 32]);
#pragma unroll
        for (int k = 16; k >= 1; k >>= 1) m = fmaxf(m, __shfl_xor(m, k, 32));
        if (tid == 0) redA[bid] = m;
      }
    }
    grid_barrier(bar, NB);

    // P4: global max, exp (scores live in LDS gsh), per-WG sum
    {
      if (tid == 0) {
        float m = redA[0];
        for (int i = 1; i < NB; ++i) m = fmaxf(m, redA[i]);
        bcast = m;
      }
      __syncthreads();
      const float gmax = bcast;
      const int s0 = bid * 64;
      if (tid < 64) {
        const float e = expf(gsh[tid] - gmax);
        wts[s0 + tid] = e;
        rsh[tid] = e;
      }
      __syncthreads();
      if (tid == 0) {
        float s = 0.f;
        for (int i = 0; i < 64; ++i) s += rsh[i];
        redB[bid] = s;
      }
    }
    grid_barrier(bar, NB);

    // P5: ctx[j] = (sum_s w_s * enc_out[s][j]) / sum   (16 j per WG)
    {
      if (tid == 0) {
        float s = 0.f;
        for (int i = 0; i < NB; ++i) s += redB[i];
        bcast = 1.f / s;
      }
      for (int i = tid; i < SEQ_E; i += TPB) sh[i] = wts[i];
      __syncthreads();
      const float inv = bcast;
      const int jj = tid & 15, chunk = tid >> 4;
      float acc = 0.f;
      const float* col = enc_out + j0 + jj;
      for (int s = chunk * 256; s < chunk * 256 + 256; ++s)
        acc = fmaf(sh[s], col[(size_t)s * HID], acc);
      rsh[tid] = acc;
      __syncthreads();
      if (tid < SLICE) {
        float v = 0.f;
#pragma unroll
        for (int c = 0; c < 16; ++c) v += rsh[c * 16 + tid];
        ctx[j0 + tid] = v * inv;
      }
    }
    grid_barrier(bar, NB);

    // P6: logits[v] = [h1;ctx] . projW[v] + projB[v]   (4 rows per WG)
    {
      const float* h1n = h1buf + nxt * HID;
      for (int i = tid; i < HID; i += TPB) { sh[i] = h1n[i]; sh[HID + i] = ctx[i]; }
      __syncthreads();
      const int v0 = bid * 4;
      const int v = wave & 3, half = wave >> 2;
      const float dp = dot1024(projW + (size_t)(v0 + v) * (2 * HID) + half * HID,
                               sh + half * HID, lane);
      if (lane == 0) gsh[wave] = dp;
      __syncthreads();
      if (tid < 4)
        out[(size_t)t * NVOC + v0 + tid] = gsh[tid] + gsh[tid + 4] + projB[v0 + tid];
    }
    // No end-of-step barrier needed: every cross-step producer/consumer pair
    // is separated by at least one of the 5 in-step barriers.
  }
}

__global__ void init_kernel(const int* __restrict__ targets, int* __restrict__ dectok,
                            float* __restrict__ hzero, unsigned* __restrict__ bar) {
  const int i = blockIdx.x * blockDim.x + threadIdx.x;
  if (i < SEQ_D) dectok[i] = (i == 0) ? START_TOK : targets[i - 1];
  if (i < HID) hzero[i] = 0.f;
  if (i < 2) bar[i] = 0u;
}

extern "C" void kernel_launch(void* const* d_in, const int* in_sizes, int n_in,
                              void* d_out, int out_size, void* d_ws, size_t ws_size,
                              hipStream_t stream) {
  (void)in_sizes; (void)n_in; (void)out_size; (void)ws_size;
  const int*   inputs  = (const int*)d_in[0];
  const int*   targets = (const int*)d_in[1];
  const float* emb     = (const float*)d_in[2];
  const float* ewih0 = (const float*)d_in[3],  *ewhh0 = (const float*)d_in[4];
  const float* ebih0 = (const float*)d_in[5],  *ebhh0 = (const float*)d_in[6];
  const float* ewih1 = (const float*)d_in[7],  *ewhh1 = (const float*)d_in[8];
  const float* ebih1 = (const float*)d_in[9],  *ebhh1 = (const float*)d_in[10];
  const float* dwih0 = (const float*)d_in[11], *dwhh0 = (const float*)d_in[12];
  const float* dbih0 = (const float*)d_in[13], *dbhh0 = (const float*)d_in[14];
  const float* dwih1 = (const float*)d_in[15], *dwhh1 = (const float*)d_in[16];
  const float* dbih1 = (const float*)d_in[17], *dbhh1 = (const float*)d_in[18];
  const float* projW = (const float*)d_in[19], *projB = (const float*)d_in[20];
  float* out = (float*)d_out;

  float* ws   = (float*)d_ws;
  float* Gi   = ws;                         // 4096*3072 (reused for all 3 GEMMs)
  float* Ye0  = Gi  + (size_t)SEQ_E * H3;   // 4096*1024
  float* Ye1  = Ye0 + (size_t)SEQ_E * HID;  // 4096*1024 (= enc_out)
  float* sm   = Ye1 + (size_t)SEQ_E * HID;
  float* h0buf = sm;        sm += 2 * HID;
  float* h1buf = sm;        sm += 2 * HID;
  float* wts   = sm;        sm += SEQ_E;
  float* redA  = sm;        sm += NB;
  float* redB  = sm;        sm += NB;
  float* ctx   = sm;        sm += HID;
  float* hzero = sm;        sm += HID;
  int*   dectok = (int*)sm; sm += SEQ_D;
  unsigned* bar = (unsigned*)sm;

  init_kernel<<<SEQ_D / TPB, TPB, 0, stream>>>(targets, dectok, hzero, bar);

  const dim3 gemmE(SEQ_E / 32, H3 / 256);
  const dim3 gemmD(SEQ_D / 32, H3 / 256);

  // Encoder layer 0: fused embedding gather + input-projection GEMM, then recurrence.
  wmma_gemm_xwT_bias<<<gemmE, TPB, 0, stream>>>(emb, inputs, ewih0, ebih0, Gi, SEQ_E);
  gru_layer_seq<<<NB, TPB, 0, stream>>>(Gi, ewhh0, ebhh0, hzero, Ye0, SEQ_E, bar);
  // Encoder layer 1.
  wmma_gemm_xwT_bias<<<gemmE, TPB, 0, stream>>>(Ye0, nullptr, ewih1, ebih1, Gi, SEQ_E);
  gru_layer_seq<<<NB, TPB, 0, stream>>>(Gi, ewhh1, ebhh1, hzero, Ye1, SEQ_E, bar);
  // Decoder layer-0 input projections (teacher-forced tokens), reusing Gi.
  wmma_gemm_xwT_bias<<<gemmD, TPB, 0, stream>>>(emb, dectok, dwih0, dbih0, Gi, SEQ_D);
  // Persistent decoder (GRU x2 + attention + projection per step).
  decoder_seq<<<NB, TPB, 0, stream>>>(Gi, dwhh0, dbhh0, dwih1, dbih1, dwhh1, dbhh1,
                                      Ye1,
                                      Ye0 + (size_t)(SEQ_E - 1) * HID,
                                      Ye1 + (size_t)(SEQ_E - 1) * HID,
                                      projW, projB,
                                      h0buf, h1buf, wts, redA, redB, ctx,
                                      out, bar);
}